// CuEquivarianceLayer_67362267070644
// MI455X (gfx1250) — compile-verified
//
#include <hip/hip_runtime.h>
#include <hip/hip_bf16.h>

typedef float v2f __attribute__((ext_vector_type(2)));
typedef float v8f __attribute__((ext_vector_type(8)));

#define D 16

// ---------------------------------------------------------------------------
// Kernel 1: fold the final dense linear into the tensor-product weight.
// Wf[uv][j] = sum_w W_tp[uv*16 + w] * weight[j*16 + w]      (Wf: [256][16])
// ---------------------------------------------------------------------------
__global__ void k_fuse_weights(const float* __restrict__ Wtp,
                               const float* __restrict__ weight,
                               float* __restrict__ Wf) {
    int idx = blockIdx.x * blockDim.x + threadIdx.x;   // 0..4095
    if (idx >= 256 * D) return;
    int uv = idx >> 4;
    int j  = idx & 15;
    float acc = 0.f;
#pragma unroll
    for (int w = 0; w < D; ++w)
        acc += Wtp[uv * D + w] * weight[j * D + w];
    Wf[uv * D + j] = acc;
}

// ---------------------------------------------------------------------------
// Kernel 2: zero the neighbor-sum buffer s[N][16].
// ---------------------------------------------------------------------------
__global__ void k_zero(float* __restrict__ p, int n) {
    int i = blockIdx.x * blockDim.x + threadIdx.x;
    if (i < n) p[i] = 0.f;
}

// ---------------------------------------------------------------------------
// Kernel 3: s[row[e]][f] += x[col[e]][f].  One thread per (edge, feature).
// 12.8M f32 atomics into a 3.2MB buffer -> resolved in L2.
// ---------------------------------------------------------------------------
__global__ void k_scatter(const float* __restrict__ x,
                          const long long* __restrict__ row,
                          const long long* __restrict__ col,
                          float* __restrict__ s, int E) {
    int gid = blockIdx.x * blockDim.x + threadIdx.x;
    int e = gid >> 4;
    int f = gid & 15;
    if (e >= E) return;
    int r = (int)row[e];
    int c = (int)col[e];
    atomicAdd(&s[r * D + f], x[c * D + f]);
}

// ---------------------------------------------------------------------------
// Kernel 4: per-node bilinear form via WMMA f32 16x16x4.
//   out[n][j] = sum_{u,v} x[n][u] * s[n][v] * Wf[u*16+v][j] + bias[j]
// One wave32 handles a tile of 16 nodes: A[16 nodes x 256] (built on the fly
// as outer products), B = Wf[256 x 16] (pre-swizzled into LDS in B-fragment
// layout), K swept in 64 chunks of 4.
//
// A-frag layout (ISA 7.12.2, 32-bit A 16x4): lane L holds row M=L&15,
//   half=L>>4: VGPR0 = K=2*half, VGPR1 = K=2*half+1.
// B-frag mirror: VGPR0 = row K=2*half (col N=L&15), VGPR1 = K=2*half+1.
// C/D: VGPR r -> row M = r + 8*half, col N = L&15.
// ---------------------------------------------------------------------------
__global__ void __launch_bounds__(256) k_bilinear_wmma(
        const float* __restrict__ x,
        const float* __restrict__ s,
        const float* __restrict__ Wf,
        const float* __restrict__ bias,
        float* __restrict__ y, int N) {
    __shared__ v2f Blds[64 * 32];          // 16 KB: B fragments for all 64 K-chunks

    const int tid  = threadIdx.x;
    const int lane = tid & 31;
    const int wave = tid >> 5;
    const int half = lane >> 4;
    const int n    = lane & 15;

    // Cooperatively swizzle Wf into B-fragment layout in LDS.
#pragma unroll
    for (int k = 0; k < 8; ++k) {
        int i  = tid + k * 256;            // 0..2047
        int t  = i >> 5;                   // K-chunk
        int l  = i & 31;
        int h  = l >> 4;
        int nn = l & 15;
        int kk = 4 * t + 2 * h;
        v2f b;
        b.x = Wf[(kk + 0) * D + nn];
        b.y = Wf[(kk + 1) * D + nn];
        Blds[i] = b;
    }
    __syncthreads();

    const float bj = bias[n];

    const int base  = (blockIdx.x * 8 + wave) * 16;
    const int node  = base + n;                       // A row handled by this lane
    const int nodec = node < N ? node : (N - 1);      // clamp: keep EXEC all-ones

    // Per-lane register copies of x[node][:] and s[node][:].
    // s is pre-shifted by 2*half so every later index is compile-time
    // (avoids dynamic register indexing / scratch spills).
    float xr[16];
    float sv[14];
#pragma unroll
    for (int u = 0; u < 16; ++u) xr[u] = x[nodec * D + u];
#pragma unroll
    for (int k = 0; k < 14; ++k) sv[k] = s[nodec * D + k + 2 * half];

    v8f acc = {};
    // K = 256 in 64 chunks of 4.  For chunk t: u = t>>2, v_even = 4*(t&3).
    // A-frag: { x[u]*s[v_even+2h], x[u]*s[v_even+2h+1] } = { xr[u]*sv[4i], xr[u]*sv[4i+1] }.
#pragma unroll
    for (int t = 0; t < 64; ++t) {
        const int u  = t >> 2;
        const int vi = 4 * (t & 3);
        v2f a;
        a.x = xr[u] * sv[vi + 0];
        a.y = xr[u] * sv[vi + 1];
        v2f b = Blds[t * 32 + lane];
        acc = __builtin_amdgcn_wmma_f32_16x16x4_f32(
            /*neg_a=*/false, a, /*neg_b=*/false, b,
            /*c_mod=*/(short)0, acc, /*reuse_a=*/false, /*reuse_b=*/false);
    }

    // Store D (C-layout) with bias; mask the tail.
#pragma unroll
    for (int r = 0; r < 8; ++r) {
        int m = base + r + 8 * half;
        if (m < N) y[m * D + n] = acc[r] + bj;
    }
}

// ---------------------------------------------------------------------------
extern "C" void kernel_launch(void* const* d_in, const int* in_sizes, int n_in,
                              void* d_out, int out_size, void* d_ws, size_t ws_size,
                              hipStream_t stream) {
    const float*     x     = (const float*)d_in[0];
    const long long* eidx  = (const long long*)d_in[1];   // int64 [2, E]
    const float*     Wtp   = (const float*)d_in[2];       // [16,16,16]
    const float*     wgt   = (const float*)d_in[3];       // [16,16]
    const float*     bias  = (const float*)d_in[4];       // [16]
    float*           y     = (float*)d_out;

    const int N = in_sizes[0] / D;
    const int E = in_sizes[1] / 2;

    const long long* row = eidx;
    const long long* col = eidx + E;

    float* s  = (float*)d_ws;              // [N,16]
    float* Wf = s + (size_t)N * D;         // [256,16]

    // 1) fold linear layer into W_tp
    k_fuse_weights<<<16, 256, 0, stream>>>(Wtp, wgt, Wf);

    // 2) zero neighbor-sum buffer
    int sElems = N * D;
    k_zero<<<(sElems + 255) / 256, 256, 0, stream>>>(s, sElems);

    // 3) scatter-add neighbor features: one thread per (edge, feature)
    long long threads = (long long)E * D;
    int sblocks = (int)((threads + 255) / 256);
    k_scatter<<<sblocks, 256, 0, stream>>>(x, row, col, s, E);

    // 4) WMMA bilinear + fused bias -> final output
    int gblocks = (N + 127) / 128;         // 8 waves/block, 16 nodes/wave
    k_bilinear_wmma<<<gblocks, 256, 0, stream>>>(x, s, Wf, bias, y, N);
}